// SimpleRNN_6098853560410
// MI455X (gfx1250) — compile-verified
//
#include <hip/hip_runtime.h>
#include <hip/hip_bf16.h>
#include <math.h>

#define T_STEPS 32768
#define D_DIM   300
#define H_DIM   100

typedef __attribute__((ext_vector_type(2))) float v2f;
typedef __attribute__((ext_vector_type(8))) float v8f;

// ---------------------------------------------------------------------------
// GEMM: C[M,N] = A[M,K] @ W[N,K]^T + b1[N] (+ b2[N])
// One wave (32 threads) per 16x16 tile of C. K stepped by 4 via
// V_WMMA_F32_16X16X4_F32.  M and K are multiples of 16 / 4; N may be ragged
// (100, 300, 400). Full-N tiles take a branch-free load->wmma fast path;
// only the last ragged column tile uses zero-masked B loads + masked stores.
// ---------------------------------------------------------------------------
__global__ __launch_bounds__(32)
void wmma_gemm_xWt(const float* __restrict__ A,
                   const float* __restrict__ W,
                   const float* __restrict__ b1,
                   const float* __restrict__ b2,
                   float* __restrict__ C,
                   int M, int N, int K)
{
    const int lane = threadIdx.x;          // 0..31
    const int m0   = blockIdx.x * 16;
    const int n0   = blockIdx.y * 16;
    const int half = lane >> 4;            // 0 or 1
    const int idx  = lane & 15;

    const int m = m0 + idx;
    const int n = n0 + idx;

    const float* __restrict__ Arow = A + (size_t)m * K;

    v8f acc = {};

    if (n0 + 16 <= N) {
        // ---------------- full tile: pure load -> wmma ----------------
        const float* __restrict__ Wrow = W + (size_t)n * K;
        for (int k0 = 0; k0 < K; k0 += 4) {
            const int kb = k0 + half * 2;
            v2f a, b;
            a.x = Arow[kb];
            a.y = Arow[kb + 1];
            b.x = Wrow[kb];
            b.y = Wrow[kb + 1];
            acc = __builtin_amdgcn_wmma_f32_16x16x4_f32(
                false, a, false, b, (short)0, acc, false, false);
        }
        float bias = b1[n];
        if (b2) bias += b2[n];
#pragma unroll
        for (int v = 0; v < 8; ++v) {
            const int mm = m0 + v + half * 8;
            C[(size_t)mm * N + n] = acc[v] + bias;
        }
    } else {
        // ---------------- ragged tile: zero-fill OOB columns ----------------
        const bool  nvalid = (n < N);
        const int   nc     = nvalid ? n : 0;
        const float nsc    = nvalid ? 1.0f : 0.0f;
        const float* __restrict__ Wrow = W + (size_t)nc * K;
        for (int k0 = 0; k0 < K; k0 += 4) {
            const int kb = k0 + half * 2;
            v2f a, b;
            a.x = Arow[kb];
            a.y = Arow[kb + 1];
            b.x = Wrow[kb]     * nsc;
            b.y = Wrow[kb + 1] * nsc;
            acc = __builtin_amdgcn_wmma_f32_16x16x4_f32(
                false, a, false, b, (short)0, acc, false, false);
        }
        if (nvalid) {
            float bias = b1[n];
            if (b2) bias += b2[n];
#pragma unroll
            for (int v = 0; v < 8; ++v) {
                const int mm = m0 + v + half * 8;
                C[(size_t)mm * N + n] = acc[v] + bias;
            }
        }
    }
}

// ---------------------------------------------------------------------------
// Sequential LSTM scan, one persistent workgroup (13 wave32s).
// Thread r (< 400) keeps W_hh row r in registers; h broadcast via LDS.
// PyTorch gate order i,f,g,o:  c = sig(f)*c + sig(i)*tanh(g); h = sig(o)*tanh(c)
// ---------------------------------------------------------------------------
__global__ __launch_bounds__(416)
void lstm_scan(const float* __restrict__ xp,    // [T, 4H] precomputed x-proj+bias
               const float* __restrict__ Whh,   // [4H, H]
               float* __restrict__ hseq,        // [T, H]
               float* __restrict__ hn,          // [H] final h
               float* __restrict__ cn,          // [H] final c
               int T)
{
    __shared__ __align__(16) float h_lds[H_DIM];
    __shared__ float gates[4 * H_DIM];

    const int r = threadIdx.x;

    float w[H_DIM];
    if (r < 4 * H_DIM) {
        const float* __restrict__ wr = Whh + (size_t)r * H_DIM;
#pragma unroll
        for (int k = 0; k < H_DIM; ++k) w[k] = wr[k];
    }

    float c = 0.0f, h = 0.0f;
    if (r < H_DIM) h_lds[r] = 0.0f;
    __syncthreads();

    for (int t = 0; t < T; ++t) {
        if (r < 4 * H_DIM) {
            float acc = xp[(size_t)t * (4 * H_DIM) + r];
            const float4* __restrict__ h4 = (const float4*)h_lds;
#pragma unroll
            for (int kk = 0; kk < H_DIM / 4; ++kk) {
                const float4 hv = h4[kk];           // same-addr DS broadcast
                acc += w[4 * kk + 0] * hv.x;
                acc += w[4 * kk + 1] * hv.y;
                acc += w[4 * kk + 2] * hv.z;
                acc += w[4 * kk + 3] * hv.w;
            }
            float g;
            if (r < 2 * H_DIM || r >= 3 * H_DIM)    // i, f, o -> sigmoid
                g = 1.0f / (1.0f + __expf(-acc));
            else                                    // g -> tanh
                g = tanhf(acc);
            gates[r] = g;
        }
        __syncthreads();    // gates complete + all h_lds reads retired
        if (r < H_DIM) {
            c = gates[H_DIM + r] * c + gates[r] * gates[2 * H_DIM + r];
            h = gates[3 * H_DIM + r] * tanhf(c);
            h_lds[r] = h;
            hseq[(size_t)t * H_DIM + r] = h;
        }
        __syncthreads();    // h_lds ready for next step
    }

    if (r < H_DIM) { hn[r] = h; cn[r] = c; }
}

// ---------------------------------------------------------------------------
extern "C" void kernel_launch(void* const* d_in, const int* in_sizes, int n_in,
                              void* d_out, int out_size, void* d_ws, size_t ws_size,
                              hipStream_t stream) {
    (void)in_sizes; (void)n_in; (void)out_size; (void)ws_size;

    const float* inputs  = (const float*)d_in[0];   // [T, D]
    const float* W_inp   = (const float*)d_in[1];   // [H, D]
    const float* b_inp   = (const float*)d_in[2];   // [H]
    const float* W_ih_l0 = (const float*)d_in[3];   // [4H, H]
    const float* W_hh_l0 = (const float*)d_in[4];   // [4H, H]
    const float* b_ih_l0 = (const float*)d_in[5];   // [4H]
    const float* b_hh_l0 = (const float*)d_in[6];   // [4H]
    const float* W_ih_l1 = (const float*)d_in[7];
    const float* W_hh_l1 = (const float*)d_in[8];
    const float* b_ih_l1 = (const float*)d_in[9];
    const float* b_hh_l1 = (const float*)d_in[10];
    const float* W_out   = (const float*)d_in[11];  // [D, H]
    const float* b_out   = (const float*)d_in[12];  // [D]

    float* out = (float*)d_out;                     // outputs [T,D] ++ h_n[2,H] ++ c_n[2,H]
    float* hn0 = out + (size_t)T_STEPS * D_DIM;
    float* hn1 = hn0 + H_DIM;
    float* cn0 = hn0 + 2 * H_DIM;
    float* cn1 = hn0 + 3 * H_DIM;

    // Workspace layout (floats):
    //   [0, T*4H)            xp buffer (layer0, then reused for layer1)
    //   [T*4H, T*4H+T*H)     x = input projection (reused as h1_seq)
    //   [.. + T*H)           h0_seq
    float* ws     = (float*)d_ws;
    float* buf_xp = ws;                                   // T*400
    float* buf_x  = ws + (size_t)T_STEPS * 4 * H_DIM;     // T*100
    float* buf_h0 = buf_x + (size_t)T_STEPS * H_DIM;      // T*100
    float* buf_h1 = buf_x;                                // reuse after xp1 built

    const dim3 blk(32);
    const dim3 g_x (T_STEPS / 16, (H_DIM     + 15) / 16); // N=100
    const dim3 g_xp(T_STEPS / 16, (4 * H_DIM + 15) / 16); // N=400
    const dim3 g_o (T_STEPS / 16, (D_DIM     + 15) / 16); // N=300

    // 1) x = inputs @ W_inp^T + b_inp            [T,100]
    wmma_gemm_xWt<<<g_x, blk, 0, stream>>>(inputs, W_inp, b_inp, nullptr,
                                           buf_x, T_STEPS, H_DIM, D_DIM);
    // 2) xp0 = x @ W_ih_l0^T + b_ih_l0 + b_hh_l0 [T,400]
    wmma_gemm_xWt<<<g_xp, blk, 0, stream>>>(buf_x, W_ih_l0, b_ih_l0, b_hh_l0,
                                            buf_xp, T_STEPS, 4 * H_DIM, H_DIM);
    // 3) layer-0 recurrence
    lstm_scan<<<1, 416, 0, stream>>>(buf_xp, W_hh_l0, buf_h0, hn0, cn0, T_STEPS);
    // 4) xp1 = h0_seq @ W_ih_l1^T + b_ih_l1 + b_hh_l1
    wmma_gemm_xWt<<<g_xp, blk, 0, stream>>>(buf_h0, W_ih_l1, b_ih_l1, b_hh_l1,
                                            buf_xp, T_STEPS, 4 * H_DIM, H_DIM);
    // 5) layer-1 recurrence
    lstm_scan<<<1, 416, 0, stream>>>(buf_xp, W_hh_l1, buf_h1, hn1, cn1, T_STEPS);
    // 6) outputs = h1_seq @ W_out^T + b_out      [T,300]
    wmma_gemm_xWt<<<g_o, blk, 0, stream>>>(buf_h1, W_out, b_out, nullptr,
                                           out, T_STEPS, D_DIM, H_DIM);
}